// HoGNet_64613488001443
// MI455X (gfx1250) — compile-verified
//
#include <hip/hip_runtime.h>
#include <math.h>

// ---------------- problem constants ----------------
constexpr int   R_    = 10;
constexpr int   B_    = 8;
constexpr int   N_    = 8192;
constexpr int   H_    = 8;
constexpr int   BN    = B_ * N_;          // 65536 nodes per ROI
constexpr int   K_    = N_ / 2;           // 4096 kept by TopK
constexpr long long EDGES = 2LL * B_ * 65536;  // 1,048,576 directed edges / ROI
constexpr int   SD_   = 19;               // super feature dim
constexpr int   HEADS = 4;
constexpr int   OH_   = 16;
constexpr int   HID1  = HEADS * OH_;      // 64
constexpr float EPS_  = 1e-5f;

// workspace offsets (floats)
constexpr long long OFF_H     = 0;                      // [R*BN, 8]
constexpr long long OFF_AGG   = OFF_H     + (long long)R_*BN*H_;
constexpr long long OFF_DEG   = OFF_AGG   + (long long)R_*BN*H_;
constexpr long long OFF_DIS   = OFF_DEG   + (long long)R_*BN;
constexpr long long OFF_SCORE = OFF_DIS   + (long long)R_*BN;
constexpr long long OFF_RES   = OFF_SCORE + (long long)R_*BN;   // [R*B, 8]
constexpr long long OFF_STATS = OFF_RES   + 640;                // sum[160] sumsq[160] mu[160] rstd[160]
constexpr long long OFF_FEATS = OFF_STATS + 640;                // [R*B, 19]

typedef float v2f __attribute__((ext_vector_type(2)));
typedef float v8f __attribute__((ext_vector_type(8)));

// ---------------- 1) GCN transform: h = x @ W via V_WMMA_F32_16X16X4_F32 ----
// A: 16 nodes x 4 features, B: W (4x8) zero-padded to 4x16, C: 16x16 (use cols 0..7)
// 4 tiles per wave: one branchless B load feeds 4 unrolled WMMAs.
__global__ __launch_bounds__(256) void k_gcn_wmma(const float* __restrict__ x,
                                                  const float* __restrict__ W,
                                                  float* __restrict__ h) {
  constexpr int TPW = 4;                         // tiles per wave (4096 % 4 == 0 -> r uniform)
  int gwave = (blockIdx.x * blockDim.x + threadIdx.x) >> 5;
  int lane  = threadIdx.x & 31;
  int tile0 = gwave * TPW;
  int r  = tile0 / (BN / 16);
  int m  = lane & 15;                            // row (A) / col (B,C)
  int hi = lane >> 4;                            // K-pair {0,1} vs {2,3}

  // branchless B operand: unconditional loads (clamped col), zero-select after
  const float* Wr = W + r * 32;                  // [4,8] row-major
  int col = m & 7;
  float w0 = Wr[(hi * 2 + 0) * 8 + col];
  float w1 = Wr[(hi * 2 + 1) * 8 + col];
  v2f b;
  b.x = (m < 8) ? w0 : 0.f;
  b.y = (m < 8) ? w1 : 0.f;

  #pragma unroll
  for (int t = 0; t < TPW; ++t) {
    long long node = (long long)(tile0 + t) * 16 + m;
    v2f a = *(const v2f*)(x + node * 4 + hi * 2);      // global_load_b64

    v8f c = {};
    c = __builtin_amdgcn_wmma_f32_16x16x4_f32(false, a, false, b, (short)0, c, false, false);

    // C/D layout: c[j] = D[M = j + 8*hi][N = m]; keep N<8
    if (m < 8) {
      long long base = (long long)(tile0 + t) * 16 * H_;
      #pragma unroll
      for (int j = 0; j < 8; ++j)
        h[base + (long long)(j + hi * 8) * H_ + m] = c[j];
    }
  }
}

// ---------------- 2) init scratch -----------------------------------------
__global__ __launch_bounds__(256) void k_init(float* __restrict__ ws) {
  long long i = (long long)blockIdx.x * 256 + threadIdx.x;
  if (i < (long long)R_ * BN * H_) ws[OFF_AGG + i] = 0.f;
  if (i < (long long)R_ * BN)      ws[OFF_DEG + i] = 1.0f;  // self-loop
  if (i < 640)                     ws[OFF_RES + i] = 0.f;
  if (i < 640)                     ws[OFF_STATS + i] = 0.f;
}

// ---------------- 3) in-degree --------------------------------------------
__global__ __launch_bounds__(256) void k_degree(const int* __restrict__ ei,
                                                float* __restrict__ deg) {
  long long e = (long long)blockIdx.x * 256 + threadIdx.x;
  if (e >= (long long)R_ * EDGES) return;
  int r = (int)(e / EDGES);
  long long k = e % EDGES;
  const int* base = ei + (long long)r * 2 * EDGES;
  __builtin_prefetch(base + EDGES + k + 4096, 0, 1);   // global_prefetch_b8
  int dst = base[EDGES + k];
  atomicAdd(&deg[(long long)r * BN + dst], 1.0f);
}

// ---------------- 4) dis = deg^-1/2 ---------------------------------------
__global__ __launch_bounds__(256) void k_rsqrt(const float* __restrict__ deg,
                                               float* __restrict__ dis) {
  int i = blockIdx.x * 256 + threadIdx.x;
  if (i < R_ * BN) dis[i] = rsqrtf(deg[i]);
}

// ---------------- 5) edge aggregation -------------------------------------
__global__ __launch_bounds__(256) void k_edge_agg(const int* __restrict__ ei,
                                                  const float* __restrict__ h,
                                                  const float* __restrict__ dis,
                                                  float* __restrict__ agg) {
  long long e = (long long)blockIdx.x * 256 + threadIdx.x;
  if (e >= (long long)R_ * EDGES) return;
  int r = (int)(e / EDGES);
  long long k = e % EDGES;
  const int* base = ei + (long long)r * 2 * EDGES;
  __builtin_prefetch(base + k + 4096, 0, 1);           // prefetch src stream
  __builtin_prefetch(base + EDGES + k + 4096, 0, 1);   // prefetch dst stream
  int s = base[k];
  int d = base[EDGES + k];
  long long sn = (long long)r * BN + s;
  long long dn = (long long)r * BN + d;
  float norm = dis[sn] * dis[dn];
  const float4* hs = (const float4*)(h + sn * H_);     // 2x global_load_b128
  float4 h0 = hs[0];
  float4 h1 = hs[1];
  float* ad = agg + dn * H_;
  atomicAdd(&ad[0], h0.x * norm);
  atomicAdd(&ad[1], h0.y * norm);
  atomicAdd(&ad[2], h0.z * norm);
  atomicAdd(&ad[3], h0.w * norm);
  atomicAdd(&ad[4], h1.x * norm);
  atomicAdd(&ad[5], h1.y * norm);
  atomicAdd(&ad[6], h1.z * norm);
  atomicAdd(&ad[7], h1.w * norm);
}

// ---------------- 6) relu(agg + h/deg + b) + per-graph sums ----------------
__global__ __launch_bounds__(256) void k_postgcn(float* __restrict__ h,
                                                 const float* __restrict__ agg,
                                                 const float* __restrict__ deg,
                                                 const float* __restrict__ gcn_b,
                                                 float* __restrict__ stats) {
  int i = blockIdx.x * 256 + threadIdx.x;           // node id over R*BN
  int tid = threadIdx.x;
  int r = i / BN;
  int b = (i % BN) / N_;
  int g = r * B_ + b;                               // block-uniform (256 | N_)
  float dinv = 1.0f / deg[i];
  float* hp = h + (long long)i * H_;
  const float* ap = agg + (long long)i * H_;
  const float* bg = gcn_b + r * H_;
  float4 hv0 = ((const float4*)hp)[0];
  float4 hv1 = ((const float4*)hp)[1];
  float4 av0 = ((const float4*)ap)[0];
  float4 av1 = ((const float4*)ap)[1];
  float v[H_] = { av0.x + hv0.x * dinv + bg[0], av0.y + hv0.y * dinv + bg[1],
                  av0.z + hv0.z * dinv + bg[2], av0.w + hv0.w * dinv + bg[3],
                  av1.x + hv1.x * dinv + bg[4], av1.y + hv1.y * dinv + bg[5],
                  av1.z + hv1.z * dinv + bg[6], av1.w + hv1.w * dinv + bg[7] };
  float ls = 0.f, lq = 0.f;
  #pragma unroll
  for (int c = 0; c < H_; ++c) {
    v[c] = fmaxf(v[c], 0.f);
    ls += v[c]; lq += v[c] * v[c];
  }
  ((float4*)hp)[0] = make_float4(v[0], v[1], v[2], v[3]);
  ((float4*)hp)[1] = make_float4(v[4], v[5], v[6], v[7]);
  __shared__ float ss[256], sq[256];
  ss[tid] = ls; sq[tid] = lq;
  __syncthreads();
  for (int s2 = 128; s2 > 0; s2 >>= 1) {
    if (tid < s2) { ss[tid] += ss[tid + s2]; sq[tid] += sq[tid + s2]; }
    __syncthreads();
  }
  if (tid == 0) {
    atomicAdd(&stats[g], ss[0]);            // sum
    atomicAdd(&stats[160 + g], sq[0]);      // sumsq
  }
}

// ---------------- 7) mu / rstd per graph ----------------------------------
__global__ void k_stats(float* __restrict__ stats) {
  int g = blockIdx.x * blockDim.x + threadIdx.x;
  if (g >= R_ * B_) return;
  float cnt = (float)N_ * (float)H_;
  float mu  = stats[g] / cnt;
  float var = stats[160 + g] / cnt - mu * mu;
  stats[320 + g] = mu;
  stats[480 + g] = rsqrtf(var + EPS_);
}

// ---------------- 8) graph-LN + topk scores + residual pool ---------------
__global__ __launch_bounds__(256) void k_norm_score(float* __restrict__ h,
                                                    const float* __restrict__ stats,
                                                    const float* __restrict__ ln_w,
                                                    const float* __restrict__ ln_b,
                                                    const float* __restrict__ topk_w,
                                                    float* __restrict__ scores,
                                                    float* __restrict__ res) {
  int i = blockIdx.x * 256 + threadIdx.x;
  int tid = threadIdx.x;
  int r = i / BN;
  int b = (i % BN) / N_;
  int g = r * B_ + b;                     // block-uniform
  float mu = stats[320 + g];
  float rs = stats[480 + g];
  const float* w  = ln_w + r * H_;
  const float* bb = ln_b + r * H_;
  const float* tk = topk_w + r * H_;
  float nw = 0.f;
  #pragma unroll
  for (int c = 0; c < H_; ++c) nw += tk[c] * tk[c];
  nw = rsqrtf(nw);
  float* hp = h + (long long)i * H_;
  float4 hv0 = ((const float4*)hp)[0];
  float4 hv1 = ((const float4*)hp)[1];
  float loc[H_] = { hv0.x, hv0.y, hv0.z, hv0.w, hv1.x, hv1.y, hv1.z, hv1.w };
  float dot = 0.f;
  #pragma unroll
  for (int c = 0; c < H_; ++c) {
    float v = (loc[c] - mu) * rs * w[c] + bb[c];
    loc[c] = v;
    dot += v * tk[c];
  }
  ((float4*)hp)[0] = make_float4(loc[0], loc[1], loc[2], loc[3]);
  ((float4*)hp)[1] = make_float4(loc[4], loc[5], loc[6], loc[7]);
  scores[i] = tanhf(dot * nw);
  // residual global_mean_pool: block-level shared reduction per channel
  __shared__ float rs8[H_];
  if (tid < H_) rs8[tid] = 0.f;
  __syncthreads();
  #pragma unroll
  for (int c = 0; c < H_; ++c) atomicAdd(&rs8[c], loc[c] * (1.0f / N_));
  __syncthreads();
  if (tid < H_) atomicAdd(&res[g * H_ + tid], rs8[tid]);
}

// ---------------- 9) TopK (radix select) + pooled feature -----------------
__device__ __forceinline__ unsigned orderKey(float f) {
  unsigned u = __float_as_uint(f);
  return (u & 0x80000000u) ? ~u : (u | 0x80000000u);  // ascending order map
}

__global__ __launch_bounds__(256) void k_topk(const float* __restrict__ h,
                                              const float* __restrict__ scores,
                                              const float* __restrict__ res,
                                              const float* __restrict__ cen,
                                              float* __restrict__ feats) {
  constexpr int PER = N_ / 256;       // 32 keys per thread
  int g = blockIdx.x;                 // r*B + b
  int tid = threadIdx.x;
  const float* sc = scores + (long long)g * N_;

  // cache ordered keys in registers: one VMEM pass instead of 34
  unsigned key[PER];
  #pragma unroll
  for (int j = 0; j < PER; ++j) key[j] = orderKey(sc[tid + j * 256]);

  __shared__ unsigned cnt;
  unsigned prefix = 0;
  unsigned kreq = (unsigned)K_;       // rank K from the top
  for (int bit = 31; bit >= 0; --bit) {
    if (tid == 0) cnt = 0;
    __syncthreads();
    unsigned bmask = 1u << bit;
    unsigned hmask = ~((bmask << 1) - 1u);   // bits [31:bit+1]
    unsigned local = 0;
    #pragma unroll
    for (int j = 0; j < PER; ++j)
      if (((key[j] & hmask) == prefix) && (key[j] & bmask)) local++;
    if (local) atomicAdd(&cnt, local);
    __syncthreads();
    unsigned c1 = cnt;
    __syncthreads();                          // protect cnt vs next reset
    if (c1 >= kreq) prefix |= bmask; else kreq -= c1;
  }
  unsigned T = prefix;                        // key of K-th largest

  __shared__ float fvs[H_];
  __shared__ unsigned cnt_gt, eq_taken;
  if (tid < H_) fvs[tid] = 0.f;
  if (tid == 0) { cnt_gt = 0; eq_taken = 0; }
  __syncthreads();
  unsigned local = 0;
  #pragma unroll
  for (int j = 0; j < PER; ++j)
    if (key[j] > T) local++;
  if (local) atomicAdd(&cnt_gt, local);
  __syncthreads();
  unsigned need_eq = (unsigned)K_ - cnt_gt;
  #pragma unroll
  for (int j = 0; j < PER; ++j) {
    unsigned u = key[j];
    bool take = (u > T);
    if (!take && u == T) take = (atomicAdd(&eq_taken, 1u) < need_eq);
    if (take) {
      int n = tid + j * 256;
      float s = sc[n];
      const float4* hp = (const float4*)(h + ((long long)g * N_ + n) * H_);
      float4 h0 = hp[0], h1 = hp[1];
      atomicAdd(&fvs[0], s * h0.x);
      atomicAdd(&fvs[1], s * h0.y);
      atomicAdd(&fvs[2], s * h0.z);
      atomicAdd(&fvs[3], s * h0.w);
      atomicAdd(&fvs[4], s * h1.x);
      atomicAdd(&fvs[5], s * h1.y);
      atomicAdd(&fvs[6], s * h1.z);
      atomicAdd(&fvs[7], s * h1.w);
    }
  }
  __syncthreads();
  float* f = feats + g * SD_;
  if (tid < H_) {
    f[tid]       = res[g * H_ + tid];
    f[H_ + tid]  = fvs[tid] * (1.0f / K_);
  }
  if (tid < 3) f[16 + tid] = cen[g * 3 + tid];
}

// ---------------- 10) super-graph GAT stack (one block) --------------------
__global__ __launch_bounds__(256) void k_super(const float* __restrict__ feats,
                                               const float* __restrict__ gat1_w,
                                               const float* __restrict__ as1,
                                               const float* __restrict__ ad1,
                                               const float* __restrict__ gat1_b,
                                               const float* __restrict__ ln1w,
                                               const float* __restrict__ ln1b,
                                               const float* __restrict__ gat2_w,
                                               const float* __restrict__ as2,
                                               const float* __restrict__ ad2,
                                               const float* __restrict__ gat2_b,
                                               const float* __restrict__ ln2w,
                                               const float* __restrict__ ln2b,
                                               const float* __restrict__ lin_w,
                                               const float* __restrict__ lin_b,
                                               float* __restrict__ out) {
  const int NB = B_ * R_;  // 80 super nodes, patient-major: node = b*R + r
  int tid = threadIdx.x;
  __shared__ float sx[NB * SD_];
  __shared__ float bufA[NB * HID1];
  __shared__ float bufB[NB * HID1];
  __shared__ float es[NB * HEADS], ed[NB * HEADS];
  __shared__ float psum[B_], psq[B_];

  // gather super_x (transpose feats[r][b] -> node b*R+r)
  for (int i = tid; i < NB * SD_; i += 256) {
    int node = i / SD_, c = i % SD_;
    int b = node / R_, r = node % R_;
    sx[i] = feats[(r * B_ + b) * SD_ + c];
  }
  __syncthreads();

  // GAT1: h = x @ W  (no bias yet)
  for (int i = tid; i < NB * HID1; i += 256) {
    int node = i / HID1, o = i % HID1;
    float acc = 0.f;
    for (int k = 0; k < SD_; ++k) acc += sx[node * SD_ + k] * gat1_w[k * HID1 + o];
    bufA[i] = acc;
  }
  __syncthreads();
  // attention logits per (node, head)
  for (int i = tid; i < NB * HEADS; i += 256) {
    int node = i / HEADS, hd = i % HEADS;
    float s = 0.f, d = 0.f;
    for (int c = 0; c < OH_; ++c) {
      float v = bufA[node * HID1 + hd * OH_ + c];
      s += v * as1[hd * OH_ + c];
      d += v * ad1[hd * OH_ + c];
    }
    es[i] = s; ed[i] = d;
  }
  __syncthreads();
  // aggregate (self loop; plus hub->node for r>0), bias, ELU
  for (int i = tid; i < NB * HID1; i += 256) {
    int node = i / HID1, o = i % HID1;
    int hd = o / OH_;
    int r = node % R_;
    float v;
    if (r == 0) {
      v = bufA[i];                                    // single self edge, alpha=1
    } else {
      int hub = node - r;
      float e_s = es[node * HEADS + hd] + ed[node * HEADS + hd];
      float e_h = es[hub * HEADS + hd] + ed[node * HEADS + hd];
      e_s = e_s > 0.f ? e_s : 0.2f * e_s;
      e_h = e_h > 0.f ? e_h : 0.2f * e_h;
      float m = fmaxf(e_s, e_h);
      float a_s = expf(e_s - m), a_h = expf(e_h - m);
      v = (a_s * bufA[i] + a_h * bufA[hub * HID1 + o]) / (a_s + a_h);
    }
    v += gat1_b[o];
    bufB[i] = v > 0.f ? v : (expf(v) - 1.0f);         // ELU
  }
  __syncthreads();
  // graph LayerNorm 1 (per patient over R*64)
  if (tid < B_) { psum[tid] = 0.f; psq[tid] = 0.f; }
  __syncthreads();
  for (int i = tid; i < NB * HID1; i += 256) {
    int b = i / (R_ * HID1);
    float v = bufB[i];
    atomicAdd(&psum[b], v);
    atomicAdd(&psq[b], v * v);
  }
  __syncthreads();
  for (int i = tid; i < NB * HID1; i += 256) {
    int b = i / (R_ * HID1), o = i % HID1;
    float mu = psum[b] / (R_ * HID1);
    float var = psq[b] / (R_ * HID1) - mu * mu;
    bufB[i] = (bufB[i] - mu) * rsqrtf(var + EPS_) * ln1w[o] + ln1b[o];
  }
  __syncthreads();

  // GAT2: h2 = x1 @ W2  (64 -> 16, 1 head)
  for (int i = tid; i < NB * OH_; i += 256) {
    int node = i / OH_, o = i % OH_;
    float acc = 0.f;
    for (int k = 0; k < HID1; ++k) acc += bufB[node * HID1 + k] * gat2_w[k * OH_ + o];
    bufA[i] = acc;
  }
  __syncthreads();
  for (int node = tid; node < NB; node += 256) {
    float s = 0.f, d = 0.f;
    for (int c = 0; c < OH_; ++c) {
      float v = bufA[node * OH_ + c];
      s += v * as2[c];
      d += v * ad2[c];
    }
    es[node] = s; ed[node] = d;
  }
  __syncthreads();
  // write x2 into bufB[0 .. NB*OH)
  for (int i = tid; i < NB * OH_; i += 256) {
    int node = i / OH_, o = i % OH_;
    int r = node % R_;
    float v;
    if (r == 0) {
      v = bufA[i];
    } else {
      int hub = node - r;
      float e_s = es[node] + ed[node];
      float e_h = es[hub] + ed[node];
      e_s = e_s > 0.f ? e_s : 0.2f * e_s;
      e_h = e_h > 0.f ? e_h : 0.2f * e_h;
      float m = fmaxf(e_s, e_h);
      float a_s = expf(e_s - m), a_h = expf(e_h - m);
      v = (a_s * bufA[i] + a_h * bufA[hub * OH_ + o]) / (a_s + a_h);
    }
    v += gat2_b[o];
    v = v > 0.f ? v : (expf(v) - 1.0f);               // ELU
    bufB[i] = v;
  }
  __syncthreads();
  // graph LayerNorm 2 (per patient over R*16)
  if (tid < B_) { psum[tid] = 0.f; psq[tid] = 0.f; }
  __syncthreads();
  for (int i = tid; i < NB * OH_; i += 256) {
    int b = i / (R_ * OH_);
    float v = bufB[i];
    atomicAdd(&psum[b], v);
    atomicAdd(&psq[b], v * v);
  }
  __syncthreads();
  for (int i = tid; i < NB * OH_; i += 256) {
    int b = i / (R_ * OH_), o = i % OH_;
    float mu = psum[b] / (R_ * OH_);
    float var = psq[b] / (R_ * OH_) - mu * mu;
    bufB[i] = (bufB[i] - mu) * rsqrtf(var + EPS_) * ln2w[o] + ln2b[o];
  }
  __syncthreads();
  // mean-pool over ROIs + final linear [16 -> 2]
  if (tid < B_ * 2) {
    int b = tid / 2, j = tid % 2;
    float acc = lin_b[j];
    for (int c = 0; c < OH_; ++c) {
      float p = 0.f;
      for (int r = 0; r < R_; ++r) p += bufB[(b * R_ + r) * OH_ + c];
      acc += (p / R_) * lin_w[c * 2 + j];
    }
    out[b * 2 + j] = acc;
  }
}

// ---------------- launcher -------------------------------------------------
extern "C" void kernel_launch(void* const* d_in, const int* in_sizes, int n_in,
                              void* d_out, int out_size, void* d_ws, size_t ws_size,
                              hipStream_t stream) {
  (void)in_sizes; (void)n_in; (void)out_size; (void)ws_size;
  const float* roi_x  = (const float*)d_in[0];
  const int*   ei     = (const int*)  d_in[1];
  const float* cen    = (const float*)d_in[2];
  const float* gcn_w  = (const float*)d_in[3];
  const float* gcn_b  = (const float*)d_in[4];
  const float* ln_w   = (const float*)d_in[5];
  const float* ln_b   = (const float*)d_in[6];
  const float* topk_w = (const float*)d_in[7];
  const float* gat1_w = (const float*)d_in[8];
  const float* as1    = (const float*)d_in[9];
  const float* ad1    = (const float*)d_in[10];
  const float* gat1_b = (const float*)d_in[11];
  const float* ln1w   = (const float*)d_in[12];
  const float* ln1b   = (const float*)d_in[13];
  const float* gat2_w = (const float*)d_in[14];
  const float* as2    = (const float*)d_in[15];
  const float* ad2    = (const float*)d_in[16];
  const float* gat2_b = (const float*)d_in[17];
  const float* ln2w   = (const float*)d_in[18];
  const float* ln2b   = (const float*)d_in[19];
  const float* lin_w  = (const float*)d_in[20];
  const float* lin_b  = (const float*)d_in[21];
  float* out = (float*)d_out;
  float* ws  = (float*)d_ws;

  float* h      = ws + OFF_H;
  float* agg    = ws + OFF_AGG;
  float* deg    = ws + OFF_DEG;
  float* dis    = ws + OFF_DIS;
  float* scores = ws + OFF_SCORE;
  float* res    = ws + OFF_RES;
  float* stats  = ws + OFF_STATS;
  float* feats  = ws + OFF_FEATS;

  const int tiles = R_ * (BN / 16);                 // 40960 tiles, 4 per wave
  k_gcn_wmma<<<tiles / (4 * 8), 256, 0, stream>>>(roi_x, gcn_w, h);

  const long long initN = (long long)R_ * BN * H_;
  k_init<<<(int)((initN + 255) / 256), 256, 0, stream>>>(ws);

  const long long edgeN = (long long)R_ * EDGES;    // 10,485,760
  k_degree<<<(int)((edgeN + 255) / 256), 256, 0, stream>>>(ei, deg);
  k_rsqrt<<<(R_ * BN + 255) / 256, 256, 0, stream>>>(deg, dis);
  k_edge_agg<<<(int)((edgeN + 255) / 256), 256, 0, stream>>>(ei, h, dis, agg);

  k_postgcn<<<(R_ * BN) / 256, 256, 0, stream>>>(h, agg, deg, gcn_b, stats);
  k_stats<<<1, 128, 0, stream>>>(stats);
  k_norm_score<<<(R_ * BN) / 256, 256, 0, stream>>>(h, stats, ln_w, ln_b, topk_w,
                                                    scores, res);
  k_topk<<<R_ * B_, 256, 0, stream>>>(h, scores, res, cen, feats);

  k_super<<<1, 256, 0, stream>>>(feats, gat1_w, as1, ad1, gat1_b, ln1w, ln1b,
                                 gat2_w, as2, ad2, gat2_b, ln2w, ln2b,
                                 lin_w, lin_b, out);
}